// CausalNCMomentumAttention_Py_58763742544958
// MI455X (gfx1250) — compile-verified
//
#include <hip/hip_runtime.h>

// ---------------------------------------------------------------------------
// Causal + non-causal linear attention (elu+1 feature map) for MI455X gfx1250.
// Chunked formulation: all heavy math is 64x64x64 fp32 GEMMs mapped onto
// V_WMMA_F32_16X16X4_F32 (wave32, 16x16 tile, k=4 per instruction).
// Problem is HBM-bandwidth bound (~335MB moved vs ~6.5 GFLOP), so fp32 WMMA
// (no precision loss) is the right precision choice on this chip.
// Raw-copy tiles (V, S_prev, KV_total) are moved global->LDS by the Tensor
// Data Mover (TDM) with LDS padding matching the 65-float pitch, overlapping
// DMA with the feat() transform loads of Q/K done by the VALU.
// ---------------------------------------------------------------------------

typedef __attribute__((ext_vector_type(2))) float v2f;
typedef __attribute__((ext_vector_type(8))) float v8f;
typedef unsigned int u32;
typedef __attribute__((ext_vector_type(4))) u32 u32x4;
typedef __attribute__((ext_vector_type(8))) int i32x8;
typedef __attribute__((ext_vector_type(4))) int i32x4;

#define N_    2
#define L_    8192
#define H_    8
#define E_    64
#define M_    64
#define CHK   64                 // chunk length
#define NC    (L_/CHK)           // 128 chunks per (n,h)
#define NH    (N_*H_)            // 16 head-batches
#define NCHUNK (NH*NC)           // 2048 total chunks
#define EPS   1e-6f
#define LDSP  65                 // padded LDS row pitch (floats)

#if __has_builtin(__builtin_amdgcn_tensor_load_to_lds) && \
    __has_builtin(__builtin_amdgcn_s_wait_tensorcnt)
#define USE_TDM 1
#else
#define USE_TDM 0
#endif

__device__ __forceinline__ float featf(float x) {
  // elu(x) + 1  ==  x+1 (x>0)  else exp(x)
  return x > 0.0f ? x + 1.0f : expf(x);
}

#if USE_TDM
// Issue one TDM 2D tile load: rows x cols f32 tile from global (row stride
// row_stride_elts elements) into LDS at lds_byte_addr, padding 1 dword after
// every 64 dwords written (matches LDSP==65 pitch). Wave-level op: EXEC is
// ignored, so call sites must be wave-uniform branches. Tracked by TENSORcnt.
__device__ __forceinline__ void tdm_load_2d(const void* gptr, u32 lds_byte_addr,
                                            u32 rows, u32 cols,
                                            u32 row_stride_elts) {
  const unsigned long long ga = (unsigned long long)(size_t)gptr;
  u32x4 g0;
  g0.x = 1u;                                           // count=1, user descriptor
  g0.y = lds_byte_addr;                                // D# lds_addr
  g0.z = (u32)(ga & 0xffffffffu);                      // global_addr[31:0]
  g0.w = (u32)((ga >> 32) & 0x01ffffffu) | (2u << 30); // global_addr[56:32] | type=2
  i32x8 g1;
  g1[0] = (int)((2u << 16)      // data_size = 4B
              | (1u << 20)      // pad_enable
              | (5u << 22)      // pad_interval: 64 dwords
              | (0u << 25));    // pad_amount: 1 dword
  g1[1] = (int)(cols << 16);    // tensor_dim0[15:0]  (bits 63:48)
  g1[2] = (int)(rows << 16);    // tensor_dim1[15:0]  (bits 95:80)
  g1[3] = (int)(cols << 16);    // tile_dim0          (bits 127:112)
  g1[4] = (int)rows;            // tile_dim1          (bits 143:128)
  g1[5] = (int)row_stride_elts; // tensor_dim0_stride[31:0] (bits 191:160)
  g1[6] = 0;
  g1[7] = 0;
  const i32x4 z4 = {0, 0, 0, 0};            // groups 2/3 unused (2D tensor)
  const i32x8 z8 = {0, 0, 0, 0, 0, 0, 0, 0};
  // clang-23 / therock arity: (g0, g1, g2, g3, extra, cpol)
  __builtin_amdgcn_tensor_load_to_lds(g0, g1, z4, z4, z8, /*cpol=*/0);
}
__device__ __forceinline__ u32 lds_addr_of(const void* p) {
  return (u32)(unsigned long long)(size_t)p;  // LDS aperture: low 32 bits = LDS offset
}
#endif

// D += X * Y over k=0..63 for one 16x16 tile (ri,ci) using f32 WMMA.
// TA: A-operand element (r,k) read as X[k*LDSP + r] (transposed) else X[r*LDSP + k]
// TB: B-operand element (k,c) read as Y[c*LDSP + k] (transposed) else Y[k*LDSP + c]
// A-matrix 16x4 per-lane layout (ISA 7.12.2): lanes 0-15 hold k=0,1 in v0,v1;
// lanes 16-31 hold k=2,3. B 4x16 mirrors C/D striping (row k in vgpr/lane-group).
template<bool TA, bool TB>
__device__ __forceinline__ void gemm_tile(const float* __restrict__ X,
                                          const float* __restrict__ Y,
                                          int ri, int ci, int lane, v8f& acc) {
  const int r  = ri * 16 + (lane & 15);
  const int c  = ci * 16 + (lane & 15);
  const int kg = (lane >> 4) * 2;
#pragma unroll
  for (int kb = 0; kb < 64; kb += 4) {
    const int k0 = kb + kg, k1 = kb + kg + 1;
    v2f a, b;
    a.x = TA ? X[k0 * LDSP + r] : X[r * LDSP + k0];
    a.y = TA ? X[k1 * LDSP + r] : X[r * LDSP + k1];
    b.x = TB ? Y[c * LDSP + k0] : Y[k0 * LDSP + c];
    b.y = TB ? Y[c * LDSP + k1] : Y[k1 * LDSP + c];
    // 8 args: (neg_a, A, neg_b, B, c_mod, C, reuse_a, reuse_b)
    acc = __builtin_amdgcn_wmma_f32_16x16x4_f32(false, a, false, b,
                                                (short)0, acc, false, false);
  }
}

// ---------------------------------------------------------------------------
// Phase A: per chunk, local KV = feat(K)^T * V  (E x M) and local ksum (E).
// ---------------------------------------------------------------------------
__global__ __launch_bounds__(256)
void la_phaseA(const float* __restrict__ Kin, const float* __restrict__ Vin,
               const float* __restrict__ mask,
               float* __restrict__ kvloc, float* __restrict__ ksloc) {
  __shared__ float Kf[CHK * LDSP];
  __shared__ float Vv[CHK * LDSP];

  const int b  = blockIdx.x;
  const int nh = b / NC, c = b % NC;
  const int n  = nh / H_, h = nh % H_;
  const int l0 = c * CHK;
  const int tid = threadIdx.x;

#if USE_TDM
  if (tid < 32) {   // wave 0 issues the DMA for the raw-copy V tile
    const float* vbase = Vin + ((size_t)(n * L_ + l0) * H_ + h) * M_;
    tdm_load_2d(vbase, lds_addr_of(Vv), CHK, M_, H_ * M_);
  }
#endif
  for (int idx = tid; idx < CHK * E_; idx += 256) {
    const int t = idx >> 6, e = idx & 63;
    const size_t gi = ((size_t)(n * L_ + l0 + t) * H_ + h) * E_ + e;
    Kf[t * LDSP + e] = featf(Kin[gi]) * mask[n * L_ + l0 + t];
#if !USE_TDM
    Vv[t * LDSP + e] = Vin[gi];                       // M_ == E_
#endif
  }
#if USE_TDM
  if (tid < 32) __builtin_amdgcn_s_wait_tensorcnt(0);
#endif
  __syncthreads();

  if (tid < E_) {                                     // local key-sum
    float s = 0.0f;
    for (int t = 0; t < CHK; ++t) s += Kf[t * LDSP + tid];
    ksloc[(size_t)b * E_ + tid] = s;
  }

  const int wave = tid >> 5, lane = tid & 31, cg = lane >> 4;
#pragma unroll
  for (int tt = 0; tt < 2; ++tt) {
    const int tile = wave * 2 + tt;
    const int ri = tile >> 2, ci = tile & 3;
    v8f acc = {};
    gemm_tile<true, false>(Kf, Vv, ri, ci, lane, acc); // KV[e][m] = sum_t K[t][e]V[t][m]
    const int col = ci * 16 + (lane & 15);
#pragma unroll
    for (int v = 0; v < 8; ++v) {
      const int row = ri * 16 + v + 8 * cg;            // C/D layout row mapping
      kvloc[(size_t)b * (E_ * M_) + row * M_ + col] = acc[v];
    }
  }
}

// ---------------------------------------------------------------------------
// Phase B: in-place exclusive prefix over the 128 chunk states per (n,h),
// plus totals for the non-causal branch. Pure streaming, trivially cheap.
// ---------------------------------------------------------------------------
__global__ __launch_bounds__(256)
void la_phaseB(float* __restrict__ kvloc, float* __restrict__ ksloc,
               float* __restrict__ kvtot, float* __restrict__ kstot) {
  const int nh  = blockIdx.x;
  const int tid = threadIdx.x;

  for (int idx = tid; idx < E_ * M_; idx += 256) {
    float acc = 0.0f;
    for (int c = 0; c < NC; ++c) {
      float* p = kvloc + (size_t)(nh * NC + c) * (E_ * M_) + idx;
      const float t = *p; *p = acc; acc += t;          // exclusive prefix
    }
    kvtot[(size_t)nh * (E_ * M_) + idx] = acc;
  }
  if (tid < E_) {
    float acc = 0.0f;
    for (int c = 0; c < NC; ++c) {
      float* p = ksloc + (size_t)(nh * NC + c) * E_ + tid;
      const float t = *p; *p = acc; acc += t;
    }
    kstot[(size_t)nh * E_ + tid] = acc;
  }
}

// ---------------------------------------------------------------------------
// Phase C: per chunk — A = Qf*Kf^T (masked), Vc = (A*V + Qf*S_prev)*Zc,
// V = (Qf*KV_total)*Z.   All three GEMMs via f32 WMMA; V/S_prev/KV_total
// tiles arrive via TDM while the VALU computes feat(Q), feat(K).
// ---------------------------------------------------------------------------
__global__ __launch_bounds__(256)
void la_phaseC(const float* __restrict__ Qin, const float* __restrict__ Kin,
               const float* __restrict__ Vin, const float* __restrict__ mask,
               const float* __restrict__ kvpre, const float* __restrict__ kspre,
               const float* __restrict__ kvtot, const float* __restrict__ kstot,
               float* __restrict__ outN, float* __restrict__ outC) {
  __shared__ float Qf [CHK * LDSP];
  __shared__ float Kf [CHK * LDSP];
  __shared__ float Vv [CHK * LDSP];
  __shared__ float Sp [E_  * LDSP];    // exclusive-prefix KV state [e][m]
  __shared__ float KVt[E_  * LDSP];    // total KV [e][m]
  __shared__ float As [CHK * LDSP];    // masked QK^T
  __shared__ float KsP[E_], KsT[E_], Zc[CHK], Zn[CHK];

  const int b  = blockIdx.x;
  const int nh = b / NC, c = b % NC;
  const int n  = nh / H_, h = nh % H_;
  const int l0 = c * CHK;
  const int tid = threadIdx.x;

#if USE_TDM
  if (tid < 32) {   // wave 0 issues the three raw-copy tile DMAs
    const float* vbase = Vin + ((size_t)(n * L_ + l0) * H_ + h) * M_;
    tdm_load_2d(vbase,                          lds_addr_of(Vv),  CHK, M_, H_ * M_);
    tdm_load_2d(kvpre + (size_t)b  * (E_ * M_), lds_addr_of(Sp),  E_,  M_, M_);
    tdm_load_2d(kvtot + (size_t)nh * (E_ * M_), lds_addr_of(KVt), E_,  M_, M_);
  }
#endif
  for (int idx = tid; idx < CHK * E_; idx += 256) {
    const int t = idx >> 6, e = idx & 63;
    const size_t gi = ((size_t)(n * L_ + l0 + t) * H_ + h) * E_ + e;
    Qf[t * LDSP + e] = featf(Qin[gi]);
    Kf[t * LDSP + e] = featf(Kin[gi]) * mask[n * L_ + l0 + t];
#if !USE_TDM
    Vv[t * LDSP + e]  = Vin[gi];
    Sp[t * LDSP + e]  = kvpre[(size_t)b * (E_ * M_) + idx];
    KVt[t * LDSP + e] = kvtot[(size_t)nh * (E_ * M_) + idx];
#endif
  }
  if (tid < E_) {
    KsP[tid] = kspre[(size_t)b * E_ + tid];
    KsT[tid] = kstot[(size_t)nh * E_ + tid];
  }
#if USE_TDM
  if (tid < 32) __builtin_amdgcn_s_wait_tensorcnt(0);
#endif
  __syncthreads();

  const int wave = tid >> 5, lane = tid & 31, cg = lane >> 4;

  // ---- GEMM 1: A = Qf * Kf^T, causal-masked (keep j <= i), store to LDS ----
#pragma unroll
  for (int tt = 0; tt < 2; ++tt) {
    const int tile = wave * 2 + tt;
    const int ri = tile >> 2, ci = tile & 3;
    const int col = ci * 16 + (lane & 15);
    if (ci > ri) {                                      // strictly-upper tile: zero
#pragma unroll
      for (int v = 0; v < 8; ++v) As[(ri * 16 + v + 8 * cg) * LDSP + col] = 0.0f;
    } else {
      v8f acc = {};
      gemm_tile<false, true>(Qf, Kf, ri, ci, lane, acc);
#pragma unroll
      for (int v = 0; v < 8; ++v) {
        const int row = ri * 16 + v + 8 * cg;
        float val = acc[v];
        if (col > row) val = 0.0f;                      // diagonal-tile mask
        As[row * LDSP + col] = val;
      }
    }
  }
  __syncthreads();

  // ---- Normalizers: Zc = 1/(q.ksum_prefix + rowsum(A) + eps), Z = 1/(q.ksum_tot + eps)
  if (tid < CHK) {
    const int i = tid;
    float rs = 0.0f, qp = 0.0f, qt = 0.0f;
    for (int j = 0; j < CHK; ++j) rs += As[i * LDSP + j];
    for (int e = 0; e < E_; ++e) {
      const float q = Qf[i * LDSP + e];
      qp += q * KsP[e];
      qt += q * KsT[e];
    }
    Zc[i] = 1.0f / (qp + rs + EPS);
    Zn[i] = 1.0f / (qt + EPS);
  }
  __syncthreads();

  // ---- GEMM 2+3: Vc = (A*V + Qf*Sp)*Zc ; V = (Qf*KVt)*Z ; store to global ----
#pragma unroll
  for (int tt = 0; tt < 2; ++tt) {
    const int tile = wave * 2 + tt;
    const int ri = tile >> 2, ci = tile & 3;
    const int col = ci * 16 + (lane & 15);
    v8f accC = {};
    gemm_tile<false, false>(As, Vv, ri, ci, lane, accC);   // intra-chunk causal
    gemm_tile<false, false>(Qf, Sp, ri, ci, lane, accC);   // inter-chunk prefix
    v8f accN = {};
    gemm_tile<false, false>(Qf, KVt, ri, ci, lane, accN);  // non-causal branch
#pragma unroll
    for (int v = 0; v < 8; ++v) {
      const int row = ri * 16 + v + 8 * cg;
      const size_t go = ((size_t)(n * L_ + l0 + row) * H_ + h) * M_ + col;
      outC[go] = accC[v] * Zc[row];
      outN[go] = accN[v] * Zn[row];
    }
  }
}

// ---------------------------------------------------------------------------
extern "C" void kernel_launch(void* const* d_in, const int* in_sizes, int n_in,
                              void* d_out, int out_size, void* d_ws, size_t ws_size,
                              hipStream_t stream) {
  const float* Q    = (const float*)d_in[0];
  const float* K    = (const float*)d_in[1];
  const float* V    = (const float*)d_in[2];
  const float* mask = (const float*)d_in[3];

  float* out  = (float*)d_out;
  float* outN = out;                                  // tuple elem 0: V (non-causal)
  float* outC = out + (size_t)N_ * L_ * H_ * M_;      // tuple elem 1: Vc (causal)

  // Workspace layout (floats): ~34.3 MB total
  float* ws    = (float*)d_ws;
  float* kvloc = ws;                                  // [2048][64][64] local->prefix KV
  float* ksloc = kvloc + (size_t)NCHUNK * E_ * M_;    // [2048][64]     local->prefix ksum
  float* kvtot = ksloc + (size_t)NCHUNK * E_;         // [16][64][64]   total KV
  float* kstot = kvtot + (size_t)NH * E_ * M_;        // [16][64]       total ksum

  la_phaseA<<<NCHUNK, 256, 0, stream>>>(K, V, mask, kvloc, ksloc);
  la_phaseB<<<NH,     256, 0, stream>>>(kvloc, ksloc, kvtot, kstot);
  la_phaseC<<<NCHUNK, 256, 0, stream>>>(Q, K, V, mask, kvloc, ksloc,
                                        kvtot, kstot, outN, outC);
}